// SpatialNumDualDescriptorPM2_13443247636942
// MI455X (gfx1250) — compile-verified
//
#include <hip/hip_runtime.h>
#include <math.h>
#include <stdint.h>

typedef float v2f __attribute__((ext_vector_type(2)));
typedef float v8f __attribute__((ext_vector_type(8)));

#define GW      193          // grid H == W
#define MDIM    64
#define NHW     192          // nH == nW
#define CHUNK   48           // k2 values per workgroup
#define PADROW  68           // padded row stride (floats): 272B = 16B-aligned
#define X2P     64           // x2 row stride (broadcast reads)
#define SLICE   (MDIM * PADROW)   // padded 64x64 slice, 4352 floats

// -------- Kernel A: cosine table ct[s][i*64+j] = cos(2*pi*s / (i*64+j+2)) --------
__global__ void build_cos_table(float* __restrict__ ct) {
    int idx = blockIdx.x * blockDim.x + threadIdx.x;
    if (idx >= NHW * MDIM * MDIM) return;
    int s = idx >> 12;            // 0..191
    int r = idx & 4095;           // i*64+j
    const float two_pi = 6.2831853071795864769f;
    ct[idx] = cosf(two_pi * (float)s / (float)(r + 2));
}

// Stage one 64x64 f32 slice (contiguous in global) into a padded LDS slice,
// via CDNA5 async DMA (GLOBAL_LOAD_ASYNC_TO_LDS_B128, tracked by ASYNCcnt).
// 1024 16B chunks; each of 128 threads issues 8.
__device__ __forceinline__ void stage_slice_async(const float* __restrict__ src,
                                                  float* dstbase, int tid) {
#pragma unroll
    for (int cc = 0; cc < 8; ++cc) {
        int c  = tid + cc * 128;        // chunk id 0..1023
        int i  = c >> 4;                // row
        int jc = (c & 15) << 2;         // col (floats), 16B aligned in padded row
        uint32_t lds   = (uint32_t)(uintptr_t)(dstbase + i * PADROW + jc);
        const float* g = src + i * MDIM + jc;
        asm volatile("global_load_async_to_lds_b128 %0, %1, off"
                     :: "v"(lds), "v"(g) : "memory");
    }
}

// -------- Kernel B: fused avgpool + WMMA GEMM + tabulated einsum --------
__global__ __launch_bounds__(128, 2)
void nk_kernel(const float* __restrict__ grid,
               const float* __restrict__ Mw,   // [64][64], y = x @ Mw^T
               const float* __restrict__ P,    // [64][64]
               const float* __restrict__ ct,   // [192][4096]
               float* __restrict__ out)        // [36864][64]
{
    __shared__ __align__(16) float x2 [CHUNK * X2P];   // post-GEMM x rows  (12 KB)
    __shared__ __align__(16) float pc1[SLICE];         // P * cos1(k1)      (17 KB)
    __shared__ __align__(16) float buf[2 * SLICE];     // x_win, then double-buffered ct2
    __shared__ __align__(16) float part[128];          // j-half partials

    const int tid  = threadIdx.x;
    const int k1   = blockIdx.y;
    const int k2_0 = blockIdx.x * CHUNK;

    // ---- Phase 1: x_win rows r=0..47 (k2 = k2_0 + r) into buf (padded) ----
    for (int idx = tid; idx < CHUNK * MDIM; idx += 128) {
        int r = idx >> 6, k = idx & 63;
        int w = k2_0 + r;
        const float* g0 = grid + ((size_t)(k1 * GW + w) * MDIM) + k;
        float v = 0.25f * (g0[0] + g0[MDIM] + g0[GW * MDIM] + g0[GW * MDIM + MDIM]);
        buf[r * PADROW + k] = v;
    }
    __syncthreads();

    // ---- Phase 2: x2[r][n] = sum_k x_win[r][k] * Mw[n][k], via V_WMMA_F32_16X16X4_F32 ----
    const int wave = tid >> 5;
    const int lane = tid & 31;
    if (wave < 3) {                       // 3 waves x (16 rows x 64 cols) tiles
        const int mt  = wave;
        const int lm  = lane & 15;        // M (A) / N (B) within tile
        const int kh  = (lane >> 4) << 1; // lanes 16-31 hold K+2,K+3
        v8f z = {0.f,0.f,0.f,0.f,0.f,0.f,0.f,0.f};
        v8f acc[4] = {z, z, z, z};
        for (int k0 = 0; k0 < MDIM; k0 += 4) {
            v2f a;
            const float* arow = &buf[(mt * 16 + lm) * PADROW + k0 + kh];
            a.x = arow[0]; a.y = arow[1];
#pragma unroll
            for (int nt = 0; nt < 4; ++nt) {
                v2f b;  // B[kk][n] = Mw[n][kk]
                const float* brow = &Mw[(nt * 16 + lm) * MDIM + k0 + kh];
                b.x = brow[0]; b.y = brow[1];
                acc[nt] = __builtin_amdgcn_wmma_f32_16x16x4_f32(
                    false, a, false, b, (short)0, acc[nt], false, false);
            }
        }
        // C/D layout: vgpr v -> row v (lanes 0-15) / row v+8 (lanes 16-31), col = lane%16
        const int moff = (lane >> 4) << 3;
#pragma unroll
        for (int nt = 0; nt < 4; ++nt)
#pragma unroll
            for (int v = 0; v < 8; ++v)
                x2[(mt * 16 + v + moff) * X2P + nt * 16 + lm] = acc[nt][v];
    }
    __syncthreads();   // buf (x_win) no longer needed -> reusable as ct2 double buffer

    // ---- Prologue: async-stage slice for pass 0 into buf half 0 (overlaps pc1 build)
    stage_slice_async(ct + (size_t)k2_0 * 4096, &buf[0], tid);

    // ---- Phase 2b: pc1[i][j] = P[i,j] * ct[k1][i*64+j]  (hoisted per k1) ----
    const float* ct1 = ct + (size_t)k1 * 4096;
    for (int idx = tid; idx < 4096; idx += 128) {
        int i = idx >> 6, j = idx & 63;
        pc1[i * PADROW + j] = P[idx] * ct1[idx];
    }

    // ---- Phase 3: 48 double-buffered passes, one k2 each ----
    const int i  = tid & 63;
    const int jh = tid >> 6;        // j-half: 0 or 1 (uniform per wave pair)
    for (int p = 0; p < CHUNK; ++p) {
        const int half = p & 1;
        if (p + 1 < CHUNK)   // stream next slice into the other half while we compute
            stage_slice_async(ct + (size_t)(k2_0 + p + 1) * 4096, &buf[(half ^ 1) * SLICE], tid);

        // Async loads retire in order per wave: <=8 outstanding means pass-p's 8 are done.
        if (p + 1 < CHUNK)
            asm volatile("s_wait_asynccnt 0x8" ::: "memory");
        else
            asm volatile("s_wait_asynccnt 0x0" ::: "memory");
        __syncthreads();    // make this wave's LDS data visible to all waves

        const float4* xr = (const float4*)&x2 [p * X2P + jh * 32];         // broadcast
        const float4* pr = (const float4*)&pc1[i * PADROW + jh * 32];
        const float4* cr = (const float4*)&buf[half * SLICE + i * PADROW + jh * 32];
        float acc = 0.f;
#pragma unroll
        for (int j4 = 0; j4 < 8; ++j4) {
            float4 xv = xr[j4], pv = pr[j4], cv = cr[j4];
            acc += xv.x * pv.x * cv.x;
            acc += xv.y * pv.y * cv.y;
            acc += xv.z * pv.z * cv.z;
            acc += xv.w * pv.w * cv.w;
        }
        part[tid] = acc;
        __syncthreads();
        if (jh == 0) {
            int b = k1 * NHW + (k2_0 + p);
            out[(size_t)b * MDIM + i] = part[tid] + part[tid + 64];
        }
        __syncthreads();    // before buf half / part are overwritten next pass
    }
}

extern "C" void kernel_launch(void* const* d_in, const int* in_sizes, int n_in,
                              void* d_out, int out_size, void* d_ws, size_t ws_size,
                              hipStream_t stream) {
    (void)in_sizes; (void)n_in; (void)out_size; (void)ws_size;
    const float* grid = (const float*)d_in[0];   // [193,193,64]
    const float* Mw   = (const float*)d_in[1];   // [64,64]
    const float* P    = (const float*)d_in[2];   // [64,64]
    float*       ct   = (float*)d_ws;            // [192][4096] cosine table (3 MB)

    int n = NHW * MDIM * MDIM;
    build_cos_table<<<(n + 255) / 256, 256, 0, stream>>>(ct);

    dim3 g(4, NHW);                               // 4 chunks of 48 k2  x  192 k1
    nk_kernel<<<g, 128, 0, stream>>>(grid, Mw, P, ct, (float*)d_out);
}